// GridCoordsDown_25056839205748
// MI455X (gfx1250) — compile-verified
//
#include <hip/hip_runtime.h>
#include <cstdint>

#define TPB 256
#define PT_BLOCKS 2048   // persistent grid for the 2M-point streaming passes

#if defined(__HIP_PLATFORM_AMD__)
#define FADD(p, v) unsafeAtomicAdd((p), (v))   // lowers to global_atomic_add_f32 (no CAS loop)
#else
#define FADD(p, v) atomicAdd((p), (v))
#endif

// ws layout: [ float4 vox[M] : sums.xyz + count.w, later mean.xyz ][ u64 best[M] ]

__global__ __launch_bounds__(TPB) void init_ws_kernel(float4* __restrict__ vox,
                                                      unsigned long long* __restrict__ best,
                                                      int M) {
    int m = blockIdx.x * blockDim.x + threadIdx.x;
    if (m < M) {
        vox[m]  = make_float4(0.f, 0.f, 0.f, 0.f);
        best[m] = 0ull;   // any real point encodes > 0, and every voxel is non-empty
    }
}

// Pass 1: per-voxel sums + counts via f32 atomics into one float4 (single L2 sector per voxel).
__global__ __launch_bounds__(TPB) void accum_kernel(const float* __restrict__ coords,
                                                    const int* __restrict__ ids,
                                                    float* __restrict__ vox, int N) {
    int stride = gridDim.x * blockDim.x;
    for (int i = blockIdx.x * blockDim.x + threadIdx.x; i < N; i += stride) {
        int nxt = i + stride;
        if (nxt < N) {   // global_prefetch_b8, RT policy: keep lines for the 2nd pass (L2-resident)
            __builtin_prefetch(&coords[3 * (size_t)nxt], 0, 3);
            __builtin_prefetch(&ids[nxt], 0, 3);
        }
        int v   = ids[i];
        float x = coords[3 * (size_t)i + 0];
        float y = coords[3 * (size_t)i + 1];
        float z = coords[3 * (size_t)i + 2];
        float* p = vox + 4 * (size_t)v;
        FADD(p + 0, x);
        FADD(p + 1, y);
        FADD(p + 2, z);
        FADD(p + 3, 1.0f);
    }
}

// Turn sums into means in place (reference: sums / max(counts, 1)).
__global__ __launch_bounds__(TPB) void finalize_avg_kernel(float4* __restrict__ vox, int M) {
    int m = blockIdx.x * blockDim.x + threadIdx.x;
    if (m < M) {
        float4 s = vox[m];
        float  k = fmaxf(s.w, 1.0f);
        vox[m] = make_float4(s.x / k, s.y / k, s.z / k, 0.f);
    }
}

// Monotone encode of an IEEE float so unsigned compare == float compare.
__device__ __forceinline__ unsigned enc_f32(float f) {
    unsigned u = __float_as_uint(f);
    return (u & 0x80000000u) ? ~u : (u | 0x80000000u);
}

// Pass 2: dist = -sqrt(||mean - c||^2); single u64 atomicMax packs (enc(dist), index):
// lexicographic max == reference's segment_max(dist) then segment_max(index) among ties.
__global__ __launch_bounds__(TPB) void select_kernel(const float* __restrict__ coords,
                                                     const int* __restrict__ ids,
                                                     const float4* __restrict__ vox,
                                                     unsigned long long* __restrict__ best,
                                                     int N) {
    int stride = gridDim.x * blockDim.x;
    for (int i = blockIdx.x * blockDim.x + threadIdx.x; i < N; i += stride) {
        int nxt = i + stride;
        if (nxt < N) {
            __builtin_prefetch(&coords[3 * (size_t)nxt], 0, 3);
            __builtin_prefetch(&ids[nxt], 0, 3);
        }
        int    v = ids[i];
        float4 a = vox[v];                 // L2-resident after pass 1
        float dx = a.x - coords[3 * (size_t)i + 0];
        float dy = a.y - coords[3 * (size_t)i + 1];
        float dz = a.z - coords[3 * (size_t)i + 2];
        float dist = -sqrtf(dx * dx + dy * dy + dz * dz);   // <= -0.0
        unsigned long long pack =
            ((unsigned long long)enc_f32(dist) << 32) | (unsigned)i;
        atomicMax(best + v, pack);         // global_atomic_max_b64
    }
}

// Emit: winner index per voxel -> gather coords, write map (indices < 2^24, exact in f32).
__global__ __launch_bounds__(TPB) void emit_kernel(const float* __restrict__ coords,
                                                   const unsigned long long* __restrict__ best,
                                                   float* __restrict__ out_coords,
                                                   float* __restrict__ out_map, int M) {
    int m = blockIdx.x * blockDim.x + threadIdx.x;
    if (m < M) {
        unsigned idx = (unsigned)(best[m] & 0xFFFFFFFFull);
        out_coords[3 * (size_t)m + 0] = coords[3 * (size_t)idx + 0];
        out_coords[3 * (size_t)m + 1] = coords[3 * (size_t)idx + 1];
        out_coords[3 * (size_t)m + 2] = coords[3 * (size_t)idx + 2];
        out_map[m] = (float)idx;
    }
}

extern "C" void kernel_launch(void* const* d_in, const int* in_sizes, int n_in,
                              void* d_out, int out_size, void* d_ws, size_t ws_size,
                              hipStream_t stream) {
    const float* coords = (const float*)d_in[0];
    const int*   ids    = (const int*)d_in[1];
    const int N = in_sizes[1];          // 2,000,000 points
    const int M = out_size / 4;         // out = [M*3 coords][M map] floats -> M = 250,000

    float4*             vox  = (float4*)d_ws;
    unsigned long long* best = (unsigned long long*)((char*)d_ws + (size_t)M * sizeof(float4));

    float* out_coords = (float*)d_out;
    float* out_map    = out_coords + (size_t)3 * M;

    const int gVox = (M + TPB - 1) / TPB;

    init_ws_kernel     <<<gVox,      TPB, 0, stream>>>(vox, best, M);
    accum_kernel       <<<PT_BLOCKS, TPB, 0, stream>>>(coords, ids, (float*)vox, N);
    finalize_avg_kernel<<<gVox,      TPB, 0, stream>>>(vox, M);
    select_kernel      <<<PT_BLOCKS, TPB, 0, stream>>>(coords, ids, vox, best, N);
    emit_kernel        <<<gVox,      TPB, 0, stream>>>(coords, best, out_coords, out_map, M);
}